// HGNNLayer_85770496901406
// MI455X (gfx1250) — compile-verified
//
#include <hip/hip_runtime.h>
#include <math.h>

typedef float v2f __attribute__((ext_vector_type(2)));
typedef float v8f __attribute__((ext_vector_type(8)));

#define DIN      128   // Din == Dout == 128 (reference shapes)
#define MT       128   // block-level M tile (8 waves x 16 rows)
#define KT       32    // K chunk staged in LDS
#define LDP      132   // plain LDS row stride for transposed-A tiles (128+4)
#define LDSP     288   // pair-interleaved row stride: 2*128 + 32 (bank-conflict free)
#define NTHREADS 256   // 8 wave32s
#define KSPLIT   5     // split-K for GEMM1 occupancy (32 E-tiles x 5 = 160 blocks)

// ---------------------------------------------------------------------------
// CDNA5 WMMA: D(16x16,f32) = A(16x4,f32) * B(4x16,f32) + C
// ---------------------------------------------------------------------------
static __device__ __forceinline__ v8f wmma4f32(v2f a, v2f b, v8f c) {
  // 8 args: (neg_a, A, neg_b, B, c_mod, C, reuse_a, reuse_b)
  return __builtin_amdgcn_wmma_f32_16x16x4_f32(false, a, false, b, (short)0, c,
                                               false, false);
}

// ---------------------------------------------------------------------------
// Pair-interleaved staging: buf[k>>1][col*2 + (k&1)], so a lane's (K,K+1)
// fragment is one aligned ds_load_b64 straight into an even VGPR pair.
// Issue phase (global->regs) is separated from commit (regs->LDS) so the
// WMMA chunk on the other buffer hides the HBM latency.
// ---------------------------------------------------------------------------
struct PairFrag { float4 r0[2], r1[2]; };

static __device__ __forceinline__ void pair_issue(PairFrag& f,
                                                  const float* __restrict__ src,
                                                  long stride, long k0, long klim,
                                                  long col0, int tid) {
#pragma unroll
  for (int it = 0; it < 2; ++it) {
    int idx = tid + it * NTHREADS;        // 0..511 : 16 k-pairs x 32 float4 cols
    int k2 = idx >> 5;
    int c  = (idx & 31) << 2;
    long ka = k0 + 2 * k2, kb = ka + 1;
    f.r0[it] = make_float4(0.f, 0.f, 0.f, 0.f);
    f.r1[it] = make_float4(0.f, 0.f, 0.f, 0.f);
    if (ka < klim) f.r0[it] = *(const float4*)(src + ka * stride + col0 + c);
    if (kb < klim) f.r1[it] = *(const float4*)(src + kb * stride + col0 + c);
  }
}

// same, with per-k-row scale (dv ∘ x for GEMM1's B operand)
static __device__ __forceinline__ void pair_issue_scaled(PairFrag& f,
                                                         const float* __restrict__ src,
                                                         const float* __restrict__ scale,
                                                         long k0, long klim, int tid) {
#pragma unroll
  for (int it = 0; it < 2; ++it) {
    int idx = tid + it * NTHREADS;
    int k2 = idx >> 5;
    int c  = (idx & 31) << 2;
    long ka = k0 + 2 * k2, kb = ka + 1;
    f.r0[it] = make_float4(0.f, 0.f, 0.f, 0.f);
    f.r1[it] = make_float4(0.f, 0.f, 0.f, 0.f);
    if (ka < klim) {
      float4 v = *(const float4*)(src + ka * (long)DIN + c);
      float s = scale[ka];
      f.r0[it] = make_float4(v.x * s, v.y * s, v.z * s, v.w * s);
    }
    if (kb < klim) {
      float4 v = *(const float4*)(src + kb * (long)DIN + c);
      float s = scale[kb];
      f.r1[it] = make_float4(v.x * s, v.y * s, v.z * s, v.w * s);
    }
  }
}

static __device__ __forceinline__ void pair_commit(float (*B)[LDSP], const PairFrag& f,
                                                   int tid) {
#pragma unroll
  for (int it = 0; it < 2; ++it) {
    int idx = tid + it * NTHREADS;
    int k2 = idx >> 5;
    int c  = (idx & 31) << 2;
    float4 lo = make_float4(f.r0[it].x, f.r1[it].x, f.r0[it].y, f.r1[it].y);
    float4 hi = make_float4(f.r0[it].z, f.r1[it].z, f.r0[it].w, f.r1[it].w);
    *(float4*)&B[k2][2 * c + 0] = lo;     // 32B-aligned (k2*288 + 8*c4)
    *(float4*)&B[k2][2 * c + 4] = hi;
  }
}

// ---------------------------------------------------------------------------
// Transposed A staging (row-major source -> A_lds[k][m]), split issue/commit.
// ---------------------------------------------------------------------------
struct TrFrag { float4 v[4]; };

static __device__ __forceinline__ void tr_issue(TrFrag& f,
                                                const float* __restrict__ src,
                                                long stride, long m0, long mlim,
                                                long k0, int tid) {
#pragma unroll
  for (int it = 0; it < 4; ++it) {
    int idx = tid + it * NTHREADS;        // 0..1023 : 128 rows x 8 float4
    int m = idx >> 3;
    int k = (idx & 7) << 2;
    f.v[it] = make_float4(0.f, 0.f, 0.f, 0.f);
    if (m0 + m < mlim) f.v[it] = *(const float4*)(src + (m0 + m) * stride + k0 + k);
  }
}

static __device__ __forceinline__ void tr_commit(float (*A)[LDP], const TrFrag& f,
                                                 int tid) {
#pragma unroll
  for (int it = 0; it < 4; ++it) {
    int idx = tid + it * NTHREADS;
    int m = idx >> 3;
    int k = (idx & 7) << 2;
    A[k + 0][m] = f.v[it].x;
    A[k + 1][m] = f.v[it].y;
    A[k + 2][m] = f.v[it].z;
    A[k + 3][m] = f.v[it].w;
  }
}

// ---------------------------------------------------------------------------
// Wave-level 16(M) x 128(N) strip per KT-chunk. Fragment layout per ISA
// 7.12.2: A 16x4: lanes 0-15 hold (K0,K1), lanes 16-31 hold (K2,K3), M=lane&15.
// ---------------------------------------------------------------------------
// variant: interleaved A + interleaved B (GEMM1)
static __device__ __forceinline__ void mma_chunk_ii(const float (*A)[LDSP],
                                                    const float (*B)[LDSP],
                                                    int mbase, int lane, v8f acc[8]) {
  const int mrow = lane & 15;
  const int hs   = lane >> 4;            // 0: (K0,K1), 1: (K2,K3)
#pragma unroll
  for (int kk = 0; kk < KT; kk += 4) {
    int k2 = (kk >> 1) + hs;
    v2f a = *(const v2f*)&A[k2][(mbase + mrow) * 2];   // one ds_load_b64
#pragma unroll
    for (int j = 0; j < 8; ++j) {
      v2f b = *(const v2f*)&B[k2][(j * 16 + mrow) * 2];
      acc[j] = wmma4f32(a, b, acc[j]);
    }
  }
}

// variant: plain transposed A [KT][LDP] + interleaved B (GEMM2/3)
static __device__ __forceinline__ void mma_chunk_pi(const float (*A)[LDP],
                                                    const float (*B)[LDSP],
                                                    int mbase, int lane, v8f acc[8]) {
  const int mrow = lane & 15;
  const int hs   = lane >> 4;
  const int kb   = hs << 1;
#pragma unroll
  for (int kk = 0; kk < KT; kk += 4) {
    v2f a;
    a.x = A[kk + kb + 0][mbase + mrow];
    a.y = A[kk + kb + 1][mbase + mrow];
    int k2 = (kk >> 1) + hs;
#pragma unroll
    for (int j = 0; j < 8; ++j) {
      v2f b = *(const v2f*)&B[k2][(j * 16 + mrow) * 2];
      acc[j] = wmma4f32(a, b, acc[j]);
    }
  }
}

// ---------------------------------------------------------------------------
// 1) Fused stats pass over H: partial row sums + col sums, deterministic.
//    grid = (E/256, ceil(N/32))
// ---------------------------------------------------------------------------
__global__ __launch_bounds__(NTHREADS) void hgnn_stats(const float* __restrict__ H,
                                                       float* __restrict__ dv_part,
                                                       float* __restrict__ de_part,
                                                       int N, int E) {
  const int tid  = threadIdx.x;
  const int col  = blockIdx.x * NTHREADS + tid;
  const long r0  = (long)blockIdx.y * 32;
  const int lane = tid & 31, wave = tid >> 5;
  __shared__ float rowpart[32][9];

  float csum = 0.f;
#pragma unroll 4
  for (int r = 0; r < 32; ++r) {
    long row = r0 + r;
    float v = (row < N && col < E) ? H[row * (long)E + col] : 0.f;
    csum += v;
#pragma unroll
    for (int off = 16; off > 0; off >>= 1) v += __shfl_xor(v, off, 32);
    if (lane == 0) rowpart[r][wave] = v;
  }
  __syncthreads();
  if (tid < 32 && r0 + tid < N) {
    float s = 0.f;
#pragma unroll
    for (int w = 0; w < 8; ++w) s += rowpart[tid][w];
    dv_part[(long)blockIdx.x * N + r0 + tid] = s;
  }
  if (col < E) de_part[(long)blockIdx.y * E + col] = csum;
}

// 2) Reduce partials; safe_pow: v>0 ? v^p : 0  (p = -1/2, -1)
__global__ __launch_bounds__(NTHREADS) void hgnn_finalize(const float* __restrict__ dv_part,
                                                          const float* __restrict__ de_part,
                                                          float* __restrict__ dv,
                                                          float* __restrict__ de,
                                                          int N, int E, int nColBlk,
                                                          int nRowBlk) {
  long i = (long)blockIdx.x * NTHREADS + threadIdx.x;
  if (i < N) {
    float s = 0.f;
    for (int b = 0; b < nColBlk; ++b) s += dv_part[(long)b * N + i];
    dv[i] = (s > 0.f) ? 1.0f / sqrtf(s) : 0.f;
  }
  if (i < E) {
    float s = 0.f;
    for (int b = 0; b < nRowBlk; ++b) s += de_part[(long)b * E + i];
    de[i] = (s > 0.f) ? 1.0f / s : 0.f;
  }
}

// ---------------------------------------------------------------------------
// 3) GEMM1: y_part[split] = H^T (dv ∘ x) over private K-ranges (no atomics).
//    grid = (E/MT, KSPLIT). Double-buffered, pair-interleaved A and B.
// ---------------------------------------------------------------------------
__global__ __launch_bounds__(NTHREADS) void hgnn_gemm1(const float* __restrict__ H,
                                                       const float* __restrict__ x,
                                                       const float* __restrict__ dv,
                                                       float* __restrict__ y_part,
                                                       int N, int E, int kper) {
  __shared__ float As[2][KT / 2][LDSP];
  __shared__ float Bs[2][KT / 2][LDSP];
  const int tid = threadIdx.x, lane = tid & 31, wave = tid >> 5;
  const int mbase = wave * 16;
  const long e0 = (long)blockIdx.x * MT;
  const long k0 = (long)blockIdx.y * kper;
  const long kend = (k0 + kper < (long)N) ? k0 + kper : (long)N;

  PairFrag fa, fb;
  pair_issue(fa, H, E, k0, kend, e0, tid);          // A[m=e][k] = H[k][e0+m]
  pair_issue_scaled(fb, x, dv, k0, kend, tid);      // B[k][d]  = dv[k]*x[k][d]
  pair_commit(As[0], fa, tid);
  pair_commit(Bs[0], fb, tid);

  v8f acc[8] = {};
  int cur = 0;
  for (long k = k0; k < kend; k += KT) {
    __syncthreads();
    const bool more = (k + KT) < kend;
    if (more) {
      pair_issue(fa, H, E, k + KT, kend, e0, tid);  // prefetch next chunk (regs)
      pair_issue_scaled(fb, x, dv, k + KT, kend, tid);
    }
    mma_chunk_ii(As[cur], Bs[cur], mbase, lane, acc);
    if (more) {
      pair_commit(As[cur ^ 1], fa, tid);
      pair_commit(Bs[cur ^ 1], fb, tid);
    }
    cur ^= 1;
  }

  const int mrow = lane & 15;
  const int rbase = mbase + (lane >> 4) * 8;        // C layout: VGPR j -> M=j / M=8+j
  float* out = y_part + (long)blockIdx.y * E * DIN;
#pragma unroll
  for (int j = 0; j < 8; ++j)
#pragma unroll
    for (int jj = 0; jj < 8; ++jj)
      out[(e0 + rbase + jj) * (long)DIN + j * 16 + mrow] = acc[j][jj];
}

// 3b) yb = de ∘ Σ_split y_part   (2 MB, L2-resident input for GEMM2)
__global__ __launch_bounds__(NTHREADS) void hgnn_reduce_y(const float* __restrict__ y_part,
                                                          const float* __restrict__ de,
                                                          float* __restrict__ yb, int E) {
  long i = (long)blockIdx.x * NTHREADS + threadIdx.x;
  if (i < (long)E * DIN) {
    float s = 0.f;
#pragma unroll
    for (int sp = 0; sp < KSPLIT; ++sp) s += y_part[(long)sp * E * DIN + i];
    yb[i] = de[i >> 7] * s;
  }
}

// 3c) Wt[k][o] = W[o][k]  (once per launch; makes GEMM3's B k-major)
__global__ __launch_bounds__(NTHREADS) void hgnn_wt(const float* __restrict__ W,
                                                    float* __restrict__ Wt) {
  int i = blockIdx.x * NTHREADS + threadIdx.x;
  if (i < DIN * DIN) {
    int o = i >> 7, k = i & 127;
    Wt[k * DIN + o] = W[o * DIN + k];
  }
}

// ---------------------------------------------------------------------------
// 4) GEMM2: z = dv ∘ (H yb),  grid = ceil(N/MT). Double-buffered.
// ---------------------------------------------------------------------------
__global__ __launch_bounds__(NTHREADS) void hgnn_gemm2(const float* __restrict__ H,
                                                       const float* __restrict__ yb,
                                                       const float* __restrict__ dv,
                                                       float* __restrict__ z,
                                                       int N, int E) {
  __shared__ float As[2][KT][LDP];
  __shared__ float Bs[2][KT / 2][LDSP];
  const int tid = threadIdx.x, lane = tid & 31, wave = tid >> 5;
  const int mbase = wave * 16;
  const long m0 = (long)blockIdx.x * MT;

  TrFrag fa;
  PairFrag fb;
  tr_issue(fa, H, E, m0, N, 0, tid);                // A[m][k] = H[m0+m][k]
  pair_issue(fb, yb, DIN, 0, E, 0, tid);            // B[k][d] = yb[k][d]
  tr_commit(As[0], fa, tid);
  pair_commit(Bs[0], fb, tid);

  v8f acc[8] = {};
  int cur = 0;
  for (long k = 0; k < E; k += KT) {
    __syncthreads();
    const bool more = (k + KT) < E;
    if (more) {
      tr_issue(fa, H, E, m0, N, k + KT, tid);
      pair_issue(fb, yb, DIN, k + KT, E, 0, tid);
    }
    mma_chunk_pi(As[cur], Bs[cur], mbase, lane, acc);
    if (more) {
      tr_commit(As[cur ^ 1], fa, tid);
      pair_commit(Bs[cur ^ 1], fb, tid);
    }
    cur ^= 1;
  }

  const int mrow = lane & 15;
  const int rbase = mbase + (lane >> 4) * 8;
#pragma unroll
  for (int j = 0; j < 8; ++j)
#pragma unroll
    for (int jj = 0; jj < 8; ++jj) {
      long row = m0 + rbase + jj;
      if (row < N) z[row * (long)DIN + j * 16 + mrow] = dv[row] * acc[j][jj];
    }
}

// ---------------------------------------------------------------------------
// 5) GEMM3: out = z Wt + b,  K = 128, grid = ceil(N/MT)
// ---------------------------------------------------------------------------
__global__ __launch_bounds__(NTHREADS) void hgnn_gemm3(const float* __restrict__ z,
                                                       const float* __restrict__ Wt,
                                                       const float* __restrict__ bias,
                                                       float* __restrict__ out, int N) {
  __shared__ float As[2][KT][LDP];
  __shared__ float Bs[2][KT / 2][LDSP];
  const int tid = threadIdx.x, lane = tid & 31, wave = tid >> 5;
  const int mbase = wave * 16;
  const long m0 = (long)blockIdx.x * MT;

  TrFrag fa;
  PairFrag fb;
  tr_issue(fa, z, DIN, m0, N, 0, tid);
  pair_issue(fb, Wt, DIN, 0, DIN, 0, tid);
  tr_commit(As[0], fa, tid);
  pair_commit(Bs[0], fb, tid);

  v8f acc[8] = {};
  int cur = 0;
  for (int k = 0; k < DIN; k += KT) {
    __syncthreads();
    const bool more = (k + KT) < DIN;
    if (more) {
      tr_issue(fa, z, DIN, m0, N, k + KT, tid);
      pair_issue(fb, Wt, DIN, k + KT, DIN, 0, tid);
    }
    mma_chunk_pi(As[cur], Bs[cur], mbase, lane, acc);
    if (more) {
      tr_commit(As[cur ^ 1], fa, tid);
      pair_commit(Bs[cur ^ 1], fb, tid);
    }
    cur ^= 1;
  }

  const int mrow = lane & 15;
  const int rbase = mbase + (lane >> 4) * 8;
#pragma unroll
  for (int j = 0; j < 8; ++j)
#pragma unroll
    for (int jj = 0; jj < 8; ++jj) {
      long row = m0 + rbase + jj;
      int col = j * 16 + mrow;
      if (row < N) out[row * (long)DIN + col] = acc[j][jj] + bias[col];
    }
}

// ---------------------------------------------------------------------------
extern "C" void kernel_launch(void* const* d_in, const int* in_sizes, int n_in,
                              void* d_out, int out_size, void* d_ws, size_t ws_size,
                              hipStream_t stream) {
  const float* x = (const float*)d_in[0];   // [N, 128]
  const float* H = (const float*)d_in[1];   // [N, E]
  const float* W = (const float*)d_in[2];   // [128, 128]
  const float* b = (const float*)d_in[3];   // [128]
  float* out = (float*)d_out;

  const int N = in_sizes[0] / DIN;          // 20000
  const int E = in_sizes[1] / N;            // 4096
  const int nColBlk = (E + NTHREADS - 1) / NTHREADS;   // 16
  const int nRowBlk = (N + 31) / 32;                   // 625

  // Workspace layout (floats), all regions 64B-aligned.
  float* ws = (float*)d_ws;
  auto a16 = [](long v) { return (v + 15) & ~15L; };
  long o = 0;
  float* dv_part = ws + o; o += a16((long)nColBlk * N);
  float* de_part = ws + o; o += a16((long)nRowBlk * E);
  float* dv      = ws + o; o += a16((long)N);
  float* de      = ws + o; o += a16((long)E);
  float* y_part  = ws + o; o += a16((long)KSPLIT * E * DIN);
  float* yb      = ws + o; o += a16((long)E * DIN);
  float* Wt      = ws + o; o += a16((long)DIN * DIN);
  float* z       = ws + o; o += a16((long)N * DIN);
  (void)ws_size;

  // 1) stats (single pass over H)
  hgnn_stats<<<dim3(nColBlk, nRowBlk), NTHREADS, 0, stream>>>(H, dv_part, de_part, N, E);

  // 2) finalize dv, de
  {
    int mx = (N > E ? N : E);
    hgnn_finalize<<<(mx + NTHREADS - 1) / NTHREADS, NTHREADS, 0, stream>>>(
        dv_part, de_part, dv, de, N, E, nColBlk, nRowBlk);
  }

  // 3) GEMM1 split-K, then fold partials (+ de scale) and transpose W
  {
    int kper = ((N + KSPLIT * KT - 1) / (KSPLIT * KT)) * KT;   // 4000
    hgnn_gemm1<<<dim3(E / MT, KSPLIT), NTHREADS, 0, stream>>>(H, x, dv, y_part, N, E, kper);
    long ny = (long)E * DIN;
    hgnn_reduce_y<<<(int)((ny + NTHREADS - 1) / NTHREADS), NTHREADS, 0, stream>>>(
        y_part, de, yb, E);
    hgnn_wt<<<(DIN * DIN + NTHREADS - 1) / NTHREADS, NTHREADS, 0, stream>>>(W, Wt);
  }

  // 4) GEMM2: z = dv ∘ (H yb)
  int mBlocks = (N + MT - 1) / MT;                              // 157
  hgnn_gemm2<<<mBlocks, NTHREADS, 0, stream>>>(H, yb, dv, z, N, E);

  // 5) GEMM3: out = z Wt + b
  hgnn_gemm3<<<mBlocks, NTHREADS, 0, stream>>>(z, Wt, b, out, N);
}